// DecoderRNN_31791347925141
// MI455X (gfx1250) — compile-verified
//
#include <hip/hip_runtime.h>

// Sizes fixed by the reference
#define BB   64
#define TT   32
#define EE   512
#define HH   512
#define VV   32000
#define G4H  2048            // 4*H
#define MS   (33 * BB)       // 2112 rows of X (prime + 32 caption steps)

typedef __attribute__((ext_vector_type(2))) float v2f;
typedef __attribute__((ext_vector_type(8))) float v8f;

__device__ __forceinline__ float sigmoidf_(float x) {
    return 1.0f / (1.0f + __expf(-x));
}

// ---------------------------------------------------------------------------
// Kernel 1: build X = [features ; embed(captions[t=0..31])]  (2112 x 512)
// ---------------------------------------------------------------------------
__global__ __launch_bounds__(256) void build_x_kernel(
    const float* __restrict__ feat,   // (64, 512)
    const int*   __restrict__ caps,   // (64, 32)
    const float* __restrict__ emb,    // (32000, 512)
    float* __restrict__ Xall)         // (2112, 512)
{
    int idx = blockIdx.x * blockDim.x + threadIdx.x;   // over MS*EE
    int row = idx >> 9;
    int col = idx & 511;
    if (row < BB) {
        Xall[idx] = feat[row * EE + col];
    } else {
        int r = row - BB;
        int t = r >> 6;          // caption step
        int b = r & 63;          // batch
        int tok = caps[b * TT + t];
        Xall[idx] = emb[(size_t)tok * EE + col];
    }
}

// ---------------------------------------------------------------------------
// Kernel 2: C[M,N] = A[M,K] @ W[N,K]^T (+bias0 +bias1), f32 WMMA 16x16x4
// block = 128 threads (4 waves); block tile = 64(M) x 256(N); K chunks of 32
// staged in LDS. Each wave: 4x4 = 16 accumulator tiles (wave owns 64 N cols,
// all 4 M-tiles) -> 4 A-frags + 4 B-frags feed 16 WMMAs per K-step, cutting
// LDS read traffic 4x vs 1-tile-per-wave.
// remap!=0 : input row r = t*64+b is stored at output row b*TT + t, and the
// store is nontemporal (write-once 262MB output must not evict W_out from L2).
// Requires: M % 64 == 0, N % 256 == 0, K % 32 == 0  (holds for all calls)
// ---------------------------------------------------------------------------
__global__ __launch_bounds__(128) void gemm64x256_nt_kernel(
    const float* __restrict__ A,
    const float* __restrict__ W,
    const float* __restrict__ bias0,
    const float* __restrict__ bias1,
    float* __restrict__ C,
    int N, int K, int remap)
{
    __shared__ float As[64][36];    // +4 pad: 9 coprime 16 -> conflict-free
    __shared__ float Bs[256][36];

    const int tid  = threadIdx.x;
    const int wave = tid >> 5;
    const int lane = tid & 31;
    const int half = lane >> 4;
    const int lm   = lane & 15;
    const int m0   = blockIdx.y * 64;
    const int n0   = blockIdx.x * 256;

    v8f acc[16];
    #pragma unroll
    for (int i = 0; i < 16; ++i) acc[i] = (v8f){};

    for (int k0 = 0; k0 < K; k0 += 32) {
        // --- stage A chunk: 64 rows x 32 (2 threads per row, 16 floats each)
        {
            int row  = tid >> 1;
            int coff = (tid & 1) * 16;
            const float* src = A + (size_t)(m0 + row) * K + k0 + coff;
            #pragma unroll
            for (int i = 0; i < 4; ++i)
                *(float4*)&As[row][coff + 4 * i] = *(const float4*)(src + 4 * i);
        }
        // --- stage W chunk: 256 rows x 32
        {
            int rb   = tid >> 1;
            int coff = (tid & 1) * 16;
            #pragma unroll
            for (int i = 0; i < 4; ++i) {
                int row = rb + i * 64;
                const float* src = W + (size_t)(n0 + row) * K + k0 + coff;
                #pragma unroll
                for (int jj = 0; jj < 4; ++jj)
                    *(float4*)&Bs[row][coff + 4 * jj] = *(const float4*)(src + 4 * jj);
            }
        }
        __syncthreads();

        #pragma unroll
        for (int kk = 0; kk < 32; kk += 4) {
            // f32 16x4 frags: lanes 0-15 hold K=kk..kk+1, lanes 16-31 K=kk+2..kk+3
            v2f a[4], b[4];
            #pragma unroll
            for (int mt = 0; mt < 4; ++mt) {
                a[mt].x = As[mt * 16 + lm][kk + 2 * half];
                a[mt].y = As[mt * 16 + lm][kk + 2 * half + 1];
            }
            #pragma unroll
            for (int nt = 0; nt < 4; ++nt) {
                int rowb = wave * 64 + nt * 16 + lm;
                b[nt].x = Bs[rowb][kk + 2 * half];
                b[nt].y = Bs[rowb][kk + 2 * half + 1];
            }
            #pragma unroll
            for (int mt = 0; mt < 4; ++mt)
                #pragma unroll
                for (int nt = 0; nt < 4; ++nt)
                    acc[mt * 4 + nt] = __builtin_amdgcn_wmma_f32_16x16x4_f32(
                        false, a[mt], false, b[nt], (short)0,
                        acc[mt * 4 + nt], false, false);
        }
        __syncthreads();
    }

    // --- epilogue: bias + (remapped, nontemporal) stores ---
    #pragma unroll
    for (int nt = 0; nt < 4; ++nt) {
        int ncol = n0 + wave * 64 + nt * 16 + lm;
        float bv = 0.0f;
        if (bias0) bv += bias0[ncol];
        if (bias1) bv += bias1[ncol];
        #pragma unroll
        for (int mt = 0; mt < 4; ++mt) {
            #pragma unroll
            for (int r = 0; r < 8; ++r) {
                int m    = m0 + mt * 16 + r + 8 * half;
                float v  = acc[mt * 4 + nt][r] + bv;
                if (remap) {
                    int orow = (m & 63) * TT + (m >> 6);
                    __builtin_nontemporal_store(v, &C[(size_t)orow * N + ncol]);
                } else {
                    C[(size_t)m * N + ncol] = v;
                }
            }
        }
    }
}

// ---------------------------------------------------------------------------
// Kernel 3: one recurrent LSTM step (fused h@W_hh^T GEMM + cell update)
// grid = 4 blocks (16 batch rows each), 256 threads = 8 waves.
// Wave w owns hidden slice [w*64, w*64+64): it computes the i/f/g/o tiles for
// that slice itself (16 accumulators), so the nonlinearity needs no exchange.
// ---------------------------------------------------------------------------
__global__ __launch_bounds__(256) void lstm_step_kernel(
    const float* __restrict__ Gx,      // (64, 2048) precomputed x-gates+biases
    const float* __restrict__ h_prev,  // (64, 512) or nullptr (== zeros)
    const float* __restrict__ c_prev,  // (64, 512) or nullptr (== zeros)
    const float* __restrict__ W_hh,    // (2048, 512)
    float* __restrict__ h_out,         // (64, 512)
    float* __restrict__ c_out)         // (64, 512)
{
    __shared__ float Hs[16][516];      // 16 x 512 h_prev tile, padded

    const int tid  = threadIdx.x;
    const int wave = tid >> 5;         // 0..7
    const int lane = tid & 31;
    const int half = lane >> 4;
    const int lm   = lane & 15;
    const int m0   = blockIdx.x * 16;  // batch tile

    v8f acc[16];
    #pragma unroll
    for (int i = 0; i < 16; ++i) acc[i] = (v8f){};

    if (h_prev) {
        // stage h_prev[m0..m0+16, :]
        int row = tid >> 4;
        int cb  = (tid & 15) * 4;
        #pragma unroll
        for (int i = 0; i < 8; ++i) {
            *(float4*)&Hs[row][cb + i * 64] =
                *(const float4*)(h_prev + (size_t)(m0 + row) * HH + cb + i * 64);
        }
        __syncthreads();

        for (int kk = 0; kk < HH; kk += 4) {
            v2f a;
            a.x = Hs[lm][kk + 2 * half];
            a.y = Hs[lm][kk + 2 * half + 1];
            #pragma unroll
            for (int i = 0; i < 16; ++i) {
                // gate (i>>2) in {i,f,g,o}, sub-tile (i&3) within wave's slice
                int nt = (i >> 2) * HH + wave * 64 + (i & 3) * 16;
                const float* wp = W_hh + (size_t)(nt + lm) * HH + kk + 2 * half;
                v2f b;
                b.x = wp[0];
                b.y = wp[1];
                acc[i] = __builtin_amdgcn_wmma_f32_16x16x4_f32(
                    false, a, false, b, (short)0, acc[i], false, false);
            }
        }
    }

    // --- cell update: c' = sig(f)*c + sig(i)*tanh(g); h' = sig(o)*tanh(c') ---
    #pragma unroll
    for (int j = 0; j < 4; ++j) {
        #pragma unroll
        for (int r = 0; r < 8; ++r) {
            int bb = m0 + r + 8 * half;              // batch row
            int hh = wave * 64 + j * 16 + lm;        // hidden col
            const float* gx = Gx + (size_t)bb * G4H;
            float iv = acc[0 * 4 + j][r] + gx[0 * HH + hh];
            float fv = acc[1 * 4 + j][r] + gx[1 * HH + hh];
            float gv = acc[2 * 4 + j][r] + gx[2 * HH + hh];
            float ov = acc[3 * 4 + j][r] + gx[3 * HH + hh];
            float cp = c_prev ? c_prev[(size_t)bb * HH + hh] : 0.0f;
            float cn = sigmoidf_(fv) * cp + sigmoidf_(iv) * tanhf(gv);
            float hn = sigmoidf_(ov) * tanhf(cn);
            c_out[(size_t)bb * HH + hh] = cn;
            h_out[(size_t)bb * HH + hh] = hn;
        }
    }
}

// ---------------------------------------------------------------------------
extern "C" void kernel_launch(void* const* d_in, const int* in_sizes, int n_in,
                              void* d_out, int out_size, void* d_ws, size_t ws_size,
                              hipStream_t stream) {
    const float* feat  = (const float*)d_in[0];
    const int*   caps  = (const int*)  d_in[1];
    // d_in[2] = seq_len (== 32, compile-time constant here)
    const float* emb   = (const float*)d_in[3];
    const float* W_ih  = (const float*)d_in[4];
    const float* W_hh  = (const float*)d_in[5];
    const float* b_ih  = (const float*)d_in[6];
    const float* b_hh  = (const float*)d_in[7];
    const float* W_out = (const float*)d_in[8];
    const float* b_out = (const float*)d_in[9];
    float* out = (float*)d_out;

    // workspace layout (floats)
    float* ws   = (float*)d_ws;
    float* Xall = ws;                              // 2112 * 512
    float* G    = Xall + (size_t)MS * EE;          // 2112 * 2048
    float* h0   = G    + (size_t)MS * G4H;         // 64 * 512
    float* Hall = h0   + (size_t)BB * HH;          // 2048 * 512 (h after steps 1..32)
    float* c0   = Hall + (size_t)TT * BB * HH;     // 64 * 512
    float* c1   = c0   + (size_t)BB * HH;          // 64 * 512

    // 1) gather inputs
    build_x_kernel<<<(MS * EE) / 256, 256, 0, stream>>>(feat, caps, emb, Xall);

    // 2) all input-side gates in one big WMMA GEMM (no recurrence):
    //    G = Xall @ W_ih^T + b_ih + b_hh       (2112 x 2048, K=512)
    gemm64x256_nt_kernel<<<dim3(G4H / 256, MS / 64), 128, 0, stream>>>(
        Xall, W_ih, b_ih, b_hh, G, G4H, EE, 0);

    // 3) sequential recurrence: step 0 primes with features, steps 1..32 scan
    for (int s = 0; s <= 32; ++s) {
        const float* hp = (s == 0) ? nullptr
                         : (s == 1) ? h0
                                    : Hall + (size_t)(s - 2) * BB * HH;
        const float* cp = (s == 0) ? nullptr : ((s & 1) ? c0 : c1);
        float* co = (s & 1) ? c1 : c0;
        float* ho = (s == 0) ? h0 : Hall + (size_t)(s - 1) * BB * HH;
        lstm_step_kernel<<<4, 256, 0, stream>>>(
            G + (size_t)s * BB * G4H, hp, cp, W_hh, ho, co);
    }

    // 4) output projection: (2048 x 32000, K=512), rows remapped (t,b)->(b,t)
    gemm64x256_nt_kernel<<<dim3(VV / 256, (TT * BB) / 64), 128, 0, stream>>>(
        Hall, W_out, b_out, nullptr, out, VV, HH, 1);
}